// SpatialMamba_59725815218624
// MI455X (gfx1250) — compile-verified
//
#include <hip/hip_runtime.h>
#include <hip/hip_bf16.h>

typedef float v2f __attribute__((ext_vector_type(2)));
typedef float v8f __attribute__((ext_vector_type(8)));

// ---------------------------------------------------------------------------
// FP32 WMMA GEMM: C[M,N] = alpha * A[M,K=64] @ B[64,N]   (all row-major)
// N is a template parameter so every B load / C store gets an immediate
// 24-bit instruction offset (no v_add_nc_u64 address chains).
// One wave computes one 16x16 C tile via 16x V_WMMA_F32_16X16X4_F32.
// Fragment layouts per CDNA5 ISA 7.12.2 (wave32):
//   A (16x4):  lanes 0-15 row M=lane,    VGPR0/1 = K=k0+0,k0+1
//              lanes 16-31 row M=lane-16, VGPR0/1 = K=k0+2,k0+3
//   B (4x16):  VGPR v, lanes 0-15: K=k0+v,   N=lane
//                       lanes 16-31: K=k0+v+2, N=lane-16
//   C (16x16): VGPR v, lanes 0-15: M=tm+v,   N=lane
//                       lanes 16-31: M=tm+v+8, N=lane-16
// Requires M%16==0, N%16==0, gridDim.x = (M/16)*(N/16)/8, block=256.
// ---------------------------------------------------------------------------
template <int N>
__global__ __launch_bounds__(256) void gemm_f32_wmma(
    const float* __restrict__ A, const float* __restrict__ B,
    float* __restrict__ C, float alpha) {
  constexpr int K = 64;
  const int lane = threadIdx.x & 31;
  const int wave = threadIdx.x >> 5;
  const int tile = blockIdx.x * 8 + wave;
  constexpr int tilesN = N >> 4;
  const int tm = (tile / tilesN) << 4;
  const int tn = (tile % tilesN) << 4;
  const int hi = lane >> 4;  // 0: lanes 0-15, 1: lanes 16-31
  const int lo = lane & 15;

  const float* Arow = A + (size_t)(tm + lo) * K + 2 * hi;  // + k0
  const float* Bcol = B + (size_t)(2 * hi) * N + tn + lo;  // + k0*N

  v8f acc = {0.f, 0.f, 0.f, 0.f, 0.f, 0.f, 0.f, 0.f};
#pragma unroll
  for (int k0 = 0; k0 < K; k0 += 4) {
    v2f a, b;
    a.x = Arow[k0];          // contiguous pair -> global_load_b64 + imm offset
    a.y = Arow[k0 + 1];
    b.x = Bcol[k0 * N];      // imm offsets (k0*N compile-time)
    b.y = Bcol[(k0 + 1) * N];
    acc = __builtin_amdgcn_wmma_f32_16x16x4_f32(
        /*neg_a=*/false, a, /*neg_b=*/false, b,
        /*c_mod=*/(short)0, acc, /*reuse_a=*/false, /*reuse_b=*/false);
  }
  float* Crow = C + (size_t)(tm + 8 * hi) * N + tn + lo;
#pragma unroll
  for (int v = 0; v < 8; ++v) Crow[v * N] = alpha * acc[v];  // imm offsets
}

// ---------------------------------------------------------------------------
// Causal depthwise conv (4 taps) along the scan direction + SiLU.
// Input: x_ssm half of xz (cols 0..63 of the 128-wide xz rows).
// dir: 0 = +H, 1 = -H, 2 = +W, 3 = -W.
// Grid: (8192*64)/256 blocks of 256 threads.
// ---------------------------------------------------------------------------
__global__ __launch_bounds__(256) void conv_silu_kernel(
    const float* __restrict__ xz, const float* __restrict__ cw,
    const float* __restrict__ cb, float* __restrict__ xssm, int dir) {
  const int idx = blockIdx.x * 256 + threadIdx.x;
  const int c = idx & 63;
  const int p = idx >> 6;           // pixel: b*1024 + h*32 + w
  const int w = p & 31;
  const int h = (p >> 5) & 31;
  const int b = p >> 10;

  float s = cb[c];
#pragma unroll
  for (int k = 0; k < 4; ++k) {
    int hh = h, ww = w;
    bool ok;
    if (dir == 0)      { hh = h - 3 + k; ok = (hh >= 0); }
    else if (dir == 1) { hh = h + 3 - k; ok = (hh < 32); }
    else if (dir == 2) { ww = w - 3 + k; ok = (ww >= 0); }
    else               { ww = w + 3 - k; ok = (ww < 32); }
    if (ok) {
      const int q = (b << 10) + (hh << 5) + ww;
      s += cw[c * 4 + k] * xz[(size_t)q * 128 + c];
    }
  }
  xssm[(size_t)p * 64 + c] = s / (1.f + __expf(-s));  // SiLU
}

// ---------------------------------------------------------------------------
// Selective scan. One block (256 threads) per sequence (1024 sequences/dir),
// L = 32 steps. Thread tid owns d = tid>>2, n in [(tid&3)*16, +16):
// 16 state registers h[d][n]. Per step, the 192-wide x_dbl row plus x_ssm/z
// rows are staged through LDS; y_d is reduced over the 4 lanes sharing d.
// Next step's x_dbl row is prefetched (global_prefetch_b8) during compute.
// dir 0 stores into acc, dirs 1..3 accumulate.
// ---------------------------------------------------------------------------
__global__ __launch_bounds__(256) void scan_kernel(
    const float* __restrict__ xdbl, const float* __restrict__ xssm,
    const float* __restrict__ xz, const float* __restrict__ A_log,
    const float* __restrict__ Dp, float* __restrict__ acc, int dir) {
  __shared__ float sDbl[192];  // [0,64): delta raw  [64,128): Bp  [128,192): Cp
  __shared__ float sX[64];
  __shared__ float sZ[64];

  const int tid = threadIdx.x;
  const int seq = blockIdx.x;          // 0..1023
  const int d = tid >> 2;              // channel 0..63
  const int nb = (tid & 3) << 4;       // state base 0/16/32/48

  // Per-thread constants
  float Aa[16];
#pragma unroll
  for (int j = 0; j < 16; ++j) Aa[j] = -__expf(A_log[d * 64 + nb + j]);
  const float Dd = Dp[d];

  float h[16];
#pragma unroll
  for (int j = 0; j < 16; ++j) h[j] = 0.f;

  // Sequence -> pixel mapping (B=8, H=32, W=32)
  const int b = seq >> 5;
  const int fixed = seq & 31;          // w for dirs 0/1, h for dirs 2/3

  auto pix = [&](int t) -> int {
    if (dir == 0)      return (b << 10) + (t << 5) + fixed;
    else if (dir == 1) return (b << 10) + ((31 - t) << 5) + fixed;
    else if (dir == 2) return (b << 10) + (fixed << 5) + t;
    else               return (b << 10) + (fixed << 5) + (31 - t);
  };

  for (int t = 0; t < 32; ++t) {
    const int p = pix(t);

    if (tid < 192) sDbl[tid] = xdbl[(size_t)p * 192 + tid];
    if (tid >= 192) sX[tid - 192] = xssm[(size_t)p * 64 + (tid - 192)];
    if (tid < 64) sZ[tid] = xz[(size_t)p * 128 + 64 + tid];

    // Hide next step's L2 latency behind the exp/FMA chain.
    if (t + 1 < 32) {
      const int pn = pix(t + 1);
      if (tid < 192) __builtin_prefetch(&xdbl[(size_t)pn * 192 + tid], 0, 3);
    }
    __syncthreads();

    float dl = sDbl[d];
    dl = (dl > 20.f) ? dl : log1pf(__expf(dl));  // softplus
    const float xt = sX[d];
    const float dx = dl * xt;

    float y = 0.f;
#pragma unroll
    for (int j = 0; j < 16; ++j) {
      const float abar = __expf(dl * Aa[j]);
      h[j] = abar * h[j] + dx * sDbl[64 + nb + j];
      y += h[j] * sDbl[128 + nb + j];
    }
    // reduce over 4 adjacent lanes sharing d (wave32)
    y += __shfl_xor(y, 1, 32);
    y += __shfl_xor(y, 2, 32);

    if ((tid & 3) == 0) {
      const float ys = y + Dd * xt;
      const float zz = sZ[d];
      const float o = ys * (zz / (1.f + __expf(-zz)));
      const size_t oi = (size_t)p * 64 + d;
      if (dir == 0) acc[oi] = o;
      else          acc[oi] += o;
    }
    __syncthreads();
  }
}

// ---------------------------------------------------------------------------
// Host-side orchestration (graph-capture safe: kernels only, all on stream).
// Workspace layout (floats): xz[8192*128] | xssm[8192*64] | xdbl[8192*192]
//                            | acc[8192*64]   -> ~14.0 MB total.
// ---------------------------------------------------------------------------
extern "C" void kernel_launch(void* const* d_in, const int* in_sizes, int n_in,
                              void* d_out, int out_size, void* d_ws,
                              size_t ws_size, hipStream_t stream) {
  const float* x      = (const float*)d_in[0];  // (8,32,32,64)
  const float* W_in   = (const float*)d_in[1];  // (64,128)
  const float* conv_w = (const float*)d_in[2];  // (64,1,4)
  const float* conv_b = (const float*)d_in[3];  // (64,)
  const float* W_x    = (const float*)d_in[4];  // (64,192)
  const float* A_log  = (const float*)d_in[5];  // (64,64)
  const float* Dp     = (const float*)d_in[6];  // (64,)
  const float* W_out  = (const float*)d_in[7];  // (64,64)
  float* out = (float*)d_out;                   // (8,32,32,64)

  float* ws   = (float*)d_ws;
  float* xz   = ws;                       // 8192*128
  float* xssm = xz + 8192 * 128;          // 8192*64
  float* xdbl = xssm + 8192 * 64;         // 8192*192
  float* acc  = xdbl + 8192 * 192;        // 8192*64

  const int M = 8192;
  // xz = x @ W_in  (N=128): (512*8)/8 = 512 blocks
  gemm_f32_wmma<128><<<(M / 16) * (128 / 16) / 8, 256, 0, stream>>>(
      x, W_in, xz, 1.0f);

  for (int dir = 0; dir < 4; ++dir) {
    conv_silu_kernel<<<(M * 64) / 256, 256, 0, stream>>>(
        xz, conv_w, conv_b, xssm, dir);
    gemm_f32_wmma<192><<<(M / 16) * (192 / 16) / 8, 256, 0, stream>>>(
        xssm, W_x, xdbl, 1.0f);
    scan_kernel<<<1024, 256, 0, stream>>>(
        xdbl, xssm, xz, A_log, Dp, acc, dir);
  }

  // out = 0.25 * acc @ W_out  (N=64): 256 blocks
  gemm_f32_wmma<64><<<(M / 16) * (64 / 16) / 8, 256, 0, stream>>>(
      acc, W_out, out, 0.25f);
}